// HHyperGNN_13778255086167
// MI455X (gfx1250) — compile-verified
//
#include <hip/hip_runtime.h>
#include <hip/hip_bf16.h>
#include <math.h>

typedef __attribute__((ext_vector_type(2))) float v2f;
typedef __attribute__((ext_vector_type(8))) float v8f;

#define N_NODES 100000
#define N_EDGES 50000
#define NNZ     1000000

// ---------------------------------------------------------------------------
// Zero fill
// ---------------------------------------------------------------------------
__global__ void k_zero(float* __restrict__ p, size_t n) {
    size_t i = (size_t)blockIdx.x * blockDim.x + threadIdx.x;
    if (i < n) p[i] = 0.0f;
}

// ---------------------------------------------------------------------------
// Per-call invariants: edge membership counts and node attention normalizer.
// cnt[e]  = #incidences with E==e          (for scatter-mean)
// asum[v] = sum_{i: V[i]==v} homo[E[i]]    (attention denominator)
// ---------------------------------------------------------------------------
__global__ void k_counts(const int* __restrict__ V, const int* __restrict__ E,
                         const float* __restrict__ homo,
                         float* __restrict__ cnt, float* __restrict__ asum) {
    int i = blockIdx.x * blockDim.x + threadIdx.x;
    if (i >= NNZ) return;
    int e = E[i], v = V[i];
    atomicAdd(&cnt[e], 1.0f);
    atomicAdd(&asum[v], homo[e]);
}

__global__ void k_att(const int* __restrict__ V, const int* __restrict__ E,
                      const float* __restrict__ homo, const float* __restrict__ asum,
                      float* __restrict__ att) {
    int i = blockIdx.x * blockDim.x + threadIdx.x;
    if (i >= NNZ) return;
    att[i] = homo[E[i]] / asum[V[i]];   // asum >= homo[E[i]] >= 0.1 > 0
}

// ---------------------------------------------------------------------------
// GEMM  Y[M,N] = X[M,K] * W[K,N]  via V_WMMA_F32_16X16X4_F32 (wave32).
// One wave computes a 16x64 tile (4 accumulators) so the A fragment is
// loaded once per K-step and reused across 4 column tiles.
//
// A 16x4 f32 layout: lanes 0-15 hold M=lane, VGPR0/1 = K=0/1;
//                    lanes 16-31 hold M=lane-16, VGPR0/1 = K=2/3.
// B 4x16 layout (row striped across lanes): lane n holds {B[kb][n],B[kb+1][n]},
//                    kb = (lane/16)*2.
// C/D 16x16: VGPR r, lane l -> row = r + 8*(l/16), col = l%16.
// ---------------------------------------------------------------------------
__global__ __launch_bounds__(32)
void k_gemm_wmma(const float* __restrict__ X, const float* __restrict__ W,
                 float* __restrict__ Y, int M, int K, int N) {
    const int lane = threadIdx.x;
    const int half = lane >> 4;     // 0 or 1
    const int l16  = lane & 15;
    const int row0 = blockIdx.x * 16;
    const int col0 = blockIdx.y * 64;

    v8f acc[4] = {v8f{}, v8f{}, v8f{}, v8f{}};

    const float* xrow = X + (size_t)(row0 + l16) * K + half * 2;

    for (int k = 0; k < K; k += 4) {
        // A fragment: {X[m][k+2h], X[m][k+2h+1]}  (8-byte aligned: k,2h even)
        v2f a = *(const v2f*)(xrow + k);
        const int kk = k + half * 2;
        const float* wrow0 = W + (size_t)kk * N;
        const float* wrow1 = W + (size_t)(kk + 1) * N;
#pragma unroll
        for (int t = 0; t < 4; ++t) {
            int col = col0 + t * 16 + l16;
            // branch-free OOB masking keeps all 32 lanes convergent (EXEC all-1s
            // is required around WMMA)
            float msk = (col < N) ? 1.0f : 0.0f;
            int cc = (col < N) ? col : (N - 1);
            v2f b;
            b.x = wrow0[cc] * msk;
            b.y = wrow1[cc] * msk;
            acc[t] = __builtin_amdgcn_wmma_f32_16x16x4_f32(
                /*neg_a=*/false, a, /*neg_b=*/false, b,
                /*c_mod=*/(short)0, acc[t],
                /*reuse_a=*/false, /*reuse_b=*/false);
        }
    }

#pragma unroll
    for (int t = 0; t < 4; ++t) {
        int col = col0 + t * 16 + l16;
        if (col < N) {
#pragma unroll
            for (int r = 0; r < 8; ++r) {
                int row = row0 + r + half * 8;
                Y[(size_t)row * N + col] = acc[t][r];
            }
        }
    }
}

// ---------------------------------------------------------------------------
// Sparse pass 1: Xe[e,:] += Y[V[i],:]  (one incidence per 64-thread block,
// float4 per thread -> fully coalesced 1KB row bursts, atomics within one
// 1KB segment)
// ---------------------------------------------------------------------------
__global__ __launch_bounds__(64)
void k_scatter_edges(const float* __restrict__ Y, const int* __restrict__ V,
                     const int* __restrict__ E, float* __restrict__ Xe, int D) {
    int i = blockIdx.x;
    int d = threadIdx.x * 4;
    if (d >= D) return;
    int v = V[i], e = E[i];
    float4 x = *(const float4*)(Y + (size_t)v * D + d);
    float* dst = Xe + (size_t)e * D + d;
    atomicAdd(dst + 0, x.x);
    atomicAdd(dst + 1, x.y);
    atomicAdd(dst + 2, x.z);
    atomicAdd(dst + 3, x.w);
}

__global__ void k_div_cnt(float* __restrict__ Xe, const float* __restrict__ cnt, int D) {
    int e = blockIdx.x;
    float inv = 1.0f / fmaxf(cnt[e], 1.0f);
    for (int d = threadIdx.x; d < D; d += blockDim.x)
        Xe[(size_t)e * D + d] *= inv;
}

// ---------------------------------------------------------------------------
// Sparse pass 2: Xv[V[i],:] += att[i] * Xe[E[i],:]
// ---------------------------------------------------------------------------
__global__ __launch_bounds__(64)
void k_scatter_nodes(const float* __restrict__ Xe, const float* __restrict__ att,
                     const int* __restrict__ V, const int* __restrict__ E,
                     float* __restrict__ Xv, int D) {
    int i = blockIdx.x;
    int d = threadIdx.x * 4;
    if (d >= D) return;
    int v = V[i], e = E[i];
    float a = att[i];
    float4 x = *(const float4*)(Xe + (size_t)e * D + d);
    float* dst = Xv + (size_t)v * D + d;
    atomicAdd(dst + 0, a * x.x);
    atomicAdd(dst + 1, a * x.y);
    atomicAdd(dst + 2, a * x.z);
    atomicAdd(dst + 3, a * x.w);
}

// ---------------------------------------------------------------------------
// Fused: out = maybe_relu( l2norm_row( Y + Xv ) ).  One wave per row,
// wave32 __shfl_xor reduction (warpSize==32 on gfx1250).
// ---------------------------------------------------------------------------
__global__ __launch_bounds__(32)
void k_finalize(const float* __restrict__ Y, const float* __restrict__ Xv,
                float* __restrict__ Xout, int D, int relu) {
    int row = blockIdx.x;
    int lane = threadIdx.x;
    float vals[8];            // D <= 256
    float ss = 0.0f;
    int nIter = (D + 31) / 32;
    for (int it = 0; it < nIter; ++it) {
        int d = lane + 32 * it;
        float v = 0.0f;
        if (d < D) v = Y[(size_t)row * D + d] + Xv[(size_t)row * D + d];
        vals[it] = v;
        ss += v * v;
    }
#pragma unroll
    for (int off = 16; off > 0; off >>= 1)
        ss += __shfl_xor(ss, off, 32);
    float norm = sqrtf(ss);
    float scale = (norm > 0.0f) ? (1.0f / norm) : 0.0f;
    for (int it = 0; it < nIter; ++it) {
        int d = lane + 32 * it;
        if (d < D) {
            float v = vals[it] * scale;
            if (relu) v = fmaxf(v, 0.0f);
            Xout[(size_t)row * D + d] = v;
        }
    }
}

// ---------------------------------------------------------------------------
extern "C" void kernel_launch(void* const* d_in, const int* in_sizes, int n_in,
                              void* d_out, int out_size, void* d_ws, size_t ws_size,
                              hipStream_t stream) {
    (void)in_sizes; (void)n_in; (void)out_size; (void)ws_size;

    const float* X    = (const float*)d_in[0];
    const int*   V    = (const int*)  d_in[1];
    const int*   E    = (const int*)  d_in[2];
    const float* homo = (const float*)d_in[3];
    const float* W1   = (const float*)d_in[4];
    const float* W2   = (const float*)d_in[5];
    const float* Wout = (const float*)d_in[6];

    const int Nn = N_NODES, Ne = N_EDGES;
    const int D = 256;

    // workspace carve-out (floats)
    float* ws   = (float*)d_ws;
    size_t off  = 0;
    float* Y    = ws + off; off += (size_t)Nn * D;   // GEMM output, reused
    float* Xe   = ws + off; off += (size_t)Ne * D;   // edge features
    float* Xv   = ws + off; off += (size_t)Nn * D;   // node scatter accum
    float* Xbuf = ws + off; off += (size_t)Nn * D;   // layer-1 output
    float* att  = ws + off; off += (size_t)NNZ;      // per-incidence attention
    float* cnt  = ws + off; off += (size_t)Ne;       // edge counts
    float* asum = ws + off; off += (size_t)Nn;       // attention denominators

    float* Zout = (float*)d_out;                       // [N, 256] (Z)
    float* Xfin = (float*)d_out + (size_t)Nn * D;      // [N, 40]

    // --- invariants (shared by all 3 layers) ---
    k_zero<<<(Ne + 255) / 256, 256, 0, stream>>>(cnt, (size_t)Ne);
    k_zero<<<(Nn + 255) / 256, 256, 0, stream>>>(asum, (size_t)Nn);
    k_counts<<<(NNZ + 255) / 256, 256, 0, stream>>>(V, E, homo, cnt, asum);
    k_att<<<(NNZ + 255) / 256, 256, 0, stream>>>(V, E, homo, asum, att);

    // --- one hypergraph conv layer ---
    auto conv = [&](const float* Xin, const float* W, int Dout, float* Xout, int relu) {
        dim3 gg(Nn / 16, (Dout + 63) / 64);
        k_gemm_wmma<<<gg, 32, 0, stream>>>(Xin, W, Y, Nn, 256, Dout);

        size_t nEe = (size_t)Ne * Dout;
        k_zero<<<(unsigned)((nEe + 255) / 256), 256, 0, stream>>>(Xe, nEe);
        k_scatter_edges<<<NNZ, 64, 0, stream>>>(Y, V, E, Xe, Dout);
        k_div_cnt<<<Ne, 256, 0, stream>>>(Xe, cnt, Dout);

        size_t nNv = (size_t)Nn * Dout;
        k_zero<<<(unsigned)((nNv + 255) / 256), 256, 0, stream>>>(Xv, nNv);
        k_scatter_nodes<<<NNZ, 64, 0, stream>>>(Xe, att, V, E, Xv, Dout);

        k_finalize<<<Nn, 32, 0, stream>>>(Y, Xv, Xout, Dout, relu);
    };

    conv(X,    W1,   256, Xbuf, 1);  // layer 1 (+relu)
    conv(Xbuf, W2,   256, Zout, 1);  // layer 2 (+relu) -> Z (first output)
    conv(Zout, Wout,  40, Xfin, 0);  // layer 3 (no relu) -> X (second output)
}